// SparseMemory_86526411145505
// MI455X (gfx1250) — compile-verified
//
#include <hip/hip_runtime.h>
#include <math.h>

// ---- problem constants (from reference) ----
#define B_   16
#define M_   32768
#define W_   64
#define S_   128
#define C_   9
#define VIS_ 1152   // C_*S_

typedef float v2f __attribute__((ext_vector_type(2)));
typedef float v4f __attribute__((ext_vector_type(4)));
typedef float v8f __attribute__((ext_vector_type(8)));

// D = A(16x4) * B(4x16) + C, f32, wave32.
// Lane layout (ISA 7.12.2): lane l: m/n = l&15, h = l>>4.
//   A frag: a[j] = A[m][2h + j]   (j = 0,1)
//   B frag: b[j] = B[2h + j][n]
//   C/D:    c[r] -> element (row r + 8h, col l&15)
static __device__ inline v8f wmma_f32_16x16x4(v2f a, v2f b, v8f c) {
  return __builtin_amdgcn_wmma_f32_16x16x4_f32(false, a, false, b, (short)0, c,
                                               false, false);
}

// ---------- 0: wide streaming copy (memory -> memory_new, usage -> usage_new)
// RT loads (source is re-read by the gather and stays resident in 192 MB L2),
// NT stores (destination is write-once; keep it from evicting the source).
__global__ void k_copy4(const v4f* __restrict__ src, v4f* __restrict__ dst,
                        int n) {
  int i = (blockIdx.x * blockDim.x + threadIdx.x) * 2;
  if (i + 1 < n) {
    v4f a = src[i];
    v4f b = src[i + 1];
    __builtin_nontemporal_store(a, &dst[i]);
    __builtin_nontemporal_store(b, &dst[i + 1]);
  } else if (i < n) {
    v4f a = src[i];
    __builtin_nontemporal_store(a, &dst[i]);
  }
}

// ---------- 1: gather visible rows + relevant usage ----------
__global__ void k_gather(const float* __restrict__ mem,
                         const float* __restrict__ usage,
                         const int* __restrict__ rp,
                         float* __restrict__ visible,
                         float* __restrict__ relevant) {
  int idx = blockIdx.x * blockDim.x + threadIdx.x;  // over B*VIS*W
  if (idx >= B_ * VIS_ * W_) return;
  int w = idx & (W_ - 1);
  int bv = idx / W_;            // b*VIS + v
  int b = bv / VIS_;
  int pos = rp[bv];
  visible[idx] = mem[(b * M_ + pos) * W_ + w];
  if (w == 0) relevant[bv] = usage[b * M_ + pos];
}

// ---------- 2: per-token least-used indicator -> isum[b,v] ----------
__global__ void k_isum(const float* __restrict__ relevant,
                       float* __restrict__ isum) {
  int t = blockIdx.x * blockDim.x + threadIdx.x;  // over B*S
  if (t >= B_ * S_) return;
  int base = t * C_;  // == b*VIS + s*C
  float mn = relevant[base];
#pragma unroll
  for (int j = 1; j < C_; ++j) mn = fminf(mn, relevant[base + j]);
#pragma unroll
  for (int j = 0; j < C_; ++j)
    isum[base + j] = (relevant[base + j] == mn) ? 1.0f : 0.0f;
}

// ---------- 3: scores[b,s,v] = sum_w keys[b,s,w]*visible[b,v,w] (WMMA) -------
// grid: (VIS/16/8, S/16, B), block 256 = 8 waves, one 16x16 tile per wave.
__global__ void k_scores(const float* __restrict__ keys,
                         const float* __restrict__ visible,
                         float* __restrict__ scores) {
  int wave = threadIdx.x >> 5;
  int lane = threadIdx.x & 31;
  int b = blockIdx.z;
  int s0 = blockIdx.y * 16;
  int v0 = (blockIdx.x * 8 + wave) * 16;
  int m = lane & 15, h = lane >> 4;
  const float* Kp = keys + b * S_ * W_;
  const float* Vp = visible + (size_t)b * VIS_ * W_;
  v8f acc = {};
  for (int k = 0; k < W_; k += 4) {
    v2f a, bb;
    a.x = Kp[(s0 + m) * W_ + k + 2 * h];
    a.y = Kp[(s0 + m) * W_ + k + 2 * h + 1];
    bb.x = Vp[(v0 + m) * W_ + k + 2 * h];      // B[k'][n], n == lane&15
    bb.y = Vp[(v0 + m) * W_ + k + 2 * h + 1];
    acc = wmma_f32_16x16x4(a, bb, acc);
  }
#pragma unroll
  for (int r = 0; r < 8; ++r)
    scores[((size_t)(b * S_ + s0 + r + 8 * h)) * VIS_ + v0 + m] = acc[r];
}

// ---------- 4: softmax over v per (b,s) row ----------
__global__ void k_softmax(float* __restrict__ sc) {
  size_t row = blockIdx.x;  // b*S + s
  float* p = sc + row * VIS_;
  __shared__ float red[256];
  int t = threadIdx.x;
  float mx = -INFINITY;
  for (int v = t; v < VIS_; v += 256) mx = fmaxf(mx, p[v]);
  red[t] = mx;
  __syncthreads();
  for (int o = 128; o > 0; o >>= 1) {
    if (t < o) red[t] = fmaxf(red[t], red[t + o]);
    __syncthreads();
  }
  mx = red[0];
  __syncthreads();
  float sum = 0.0f;
  for (int v = t; v < VIS_; v += 256) {
    float e = __expf(p[v] - mx);
    p[v] = e;
    sum += e;
  }
  red[t] = sum;
  __syncthreads();
  for (int o = 128; o > 0; o >>= 1) {
    if (t < o) red[t] += red[t + o];
    __syncthreads();
  }
  float inv = 1.0f / red[0];
  for (int v = t; v < VIS_; v += 256) p[v] *= inv;
}

// ---------- 5: read_vectors = read_weights @ visible (WMMA, K=VIS) ----------
// grid: (W/16, S/16, B), block 32 (one wave per 16x16 tile).
__global__ void k_readvec(const float* __restrict__ rwt,
                          const float* __restrict__ visible,
                          float* __restrict__ rvec) {
  int lane = threadIdx.x & 31;
  int b = blockIdx.z;
  int s0 = blockIdx.y * 16;
  int w0 = blockIdx.x * 16;
  int m = lane & 15, h = lane >> 4;
  const float* Rp = rwt + (size_t)b * S_ * VIS_;
  const float* Vp = visible + (size_t)b * VIS_ * W_;
  v8f acc = {};
  for (int k = 0; k < VIS_; k += 4) {
    v2f a, bb;
    a.x = Rp[(s0 + m) * VIS_ + k + 2 * h];
    a.y = Rp[(s0 + m) * VIS_ + k + 2 * h + 1];
    bb.x = Vp[(k + 2 * h) * W_ + w0 + m];
    bb.y = Vp[(k + 2 * h + 1) * W_ + w0 + m];
    acc = wmma_f32_16x16x4(a, bb, acc);
  }
#pragma unroll
  for (int r = 0; r < 8; ++r)
    rvec[(b * S_ + s0 + r + 8 * h) * W_ + w0 + m] = acc[r];
}

// ---------- 6: ww[b,s,v] elementwise ----------
__global__ void k_ww(const float* __restrict__ rwt,
                     const float* __restrict__ isum,
                     const float* __restrict__ ig,
                     const float* __restrict__ wg,
                     const int* __restrict__ am,
                     const int* __restrict__ rp, float* __restrict__ ww) {
  size_t idx = (size_t)blockIdx.x * blockDim.x + threadIdx.x;  // b,s,v
  if (idx >= (size_t)B_ * S_ * VIS_) return;
  int v = (int)(idx % VIS_);
  int bs = (int)(idx / VIS_);
  int s = bs & (S_ - 1);
  int b = bs / S_;
  float is = isum[b * VIS_ + v];
  float r = rwt[idx] * (is - 1.0f);
  float Iv = (v / C_ == s) ? is : 0.0f;
  float g = ig[bs];
  float w = wg[bs] * (g * r + (1.0f - g) * Iv);
  w *= (float)am[bs];
  w *= (rp[b * VIS_ + v] != 0) ? 1.0f : 0.0f;
  ww[idx] = w;
}

// ---------- 7: activity column-sum + usage scatter ----------
__global__ void k_usage(const float* __restrict__ rwt,
                        const float* __restrict__ ww,
                        const float* __restrict__ isum,
                        const float* __restrict__ relevant,
                        const int* __restrict__ rp,
                        float* __restrict__ usage_out) {
  int idx = blockIdx.x * blockDim.x + threadIdx.x;  // b*VIS + v
  if (idx >= B_ * VIS_) return;
  int v = idx % VIS_;
  int b = idx / VIS_;
  float is = isum[idx];
  float acc = 0.0f;
  for (int s = 0; s < S_; ++s) {
    size_t o = (size_t)(b * S_ + s) * VIS_ + v;
    acc += fabsf(rwt[o] * (is - 1.0f)) + fabsf(ww[o]);
  }
  usage_out[b * M_ + rp[idx]] = relevant[idx] + acc;  // .set semantics (races OK)
}

// ---------- 8: writings = ww^T @ write_vector (WMMA, K=S) + erase/scatter ----
// grid: (VIS/16, W/16, B), block 32.
__global__ void k_write(const float* __restrict__ ww,
                        const float* __restrict__ wvec,
                        const float* __restrict__ visible,
                        const float* __restrict__ isum,
                        const int* __restrict__ rp,
                        float* __restrict__ mem_out) {
  int lane = threadIdx.x & 31;
  int b = blockIdx.z;
  int v0 = blockIdx.x * 16;
  int w0 = blockIdx.y * 16;
  int m = lane & 15, h = lane >> 4;
  const float* Wp = ww + (size_t)b * S_ * VIS_;
  const float* Bp = wvec + b * S_ * W_;
  v8f acc = {};
  for (int k = 0; k < S_; k += 4) {
    v2f a, bb;
    a.x = Wp[(size_t)(k + 2 * h) * VIS_ + v0 + m];      // A[v][s] = ww[s][v]
    a.y = Wp[(size_t)(k + 2 * h + 1) * VIS_ + v0 + m];
    bb.x = Bp[(k + 2 * h) * W_ + w0 + m];
    bb.y = Bp[(k + 2 * h + 1) * W_ + w0 + m];
    acc = wmma_f32_16x16x4(a, bb, acc);
  }
#pragma unroll
  for (int r = 0; r < 8; ++r) {
    int vv = v0 + r + 8 * h;
    float keep = 1.0f - isum[b * VIS_ + vv];
    float val = visible[((size_t)b * VIS_ + vv) * W_ + w0 + m] * keep + acc[r];
    int pos = rp[b * VIS_ + vv];
    mem_out[(b * M_ + pos) * W_ + w0 + m] = val;
  }
}

// ---------- 9: per-batch top-128 least used (iterative argmin in LDS) -------
__global__ void __launch_bounds__(1024) k_topk(const float* __restrict__ un,
                                               float* __restrict__ out_idx) {
  int b = blockIdx.x;
  __shared__ float vals[M_];      // 128 KB (fits in 320 KB WGP LDS)
  __shared__ float rv[1024];
  __shared__ int ri[1024];
  int t = threadIdx.x;
  for (int i = t; i < M_; i += 1024) vals[i] = un[b * M_ + i];
  __syncthreads();
  for (int k = 0; k < S_; ++k) {
    float bv = INFINITY;
    int bi = 0x7fffffff;
    for (int i = t; i < M_; i += 1024) {
      float x = vals[i];
      if (x < bv) { bv = x; bi = i; }
    }
    rv[t] = bv;
    ri[t] = bi;
    __syncthreads();
    for (int o = 512; o > 0; o >>= 1) {
      if (t < o) {
        if (rv[t + o] < rv[t] || (rv[t + o] == rv[t] && ri[t + o] < ri[t])) {
          rv[t] = rv[t + o];
          ri[t] = ri[t + o];
        }
      }
      __syncthreads();
    }
    if (t == 0) {
      out_idx[b * S_ + k] = (float)ri[0];
      vals[ri[0]] = INFINITY;
    }
    __syncthreads();
  }
}

extern "C" void kernel_launch(void* const* d_in, const int* in_sizes, int n_in,
                              void* d_out, int out_size, void* d_ws,
                              size_t ws_size, hipStream_t stream) {
  const float* memory = (const float*)d_in[0];
  const float* usage = (const float*)d_in[1];
  const float* keys = (const float*)d_in[2];
  const float* wvec = (const float*)d_in[3];
  const float* ig = (const float*)d_in[4];
  const float* wg = (const float*)d_in[5];
  const int* rp = (const int*)d_in[6];
  const int* am = (const int*)d_in[7];

  float* out = (float*)d_out;
  float* out_rvec = out;                                     // B*S*W
  float* out_mem = out_rvec + (size_t)B_ * S_ * W_;          // B*M*W
  float* out_usage = out_mem + (size_t)B_ * M_ * W_;         // B*M
  float* out_lum = out_usage + (size_t)B_ * M_;              // B*S

  float* visible = (float*)d_ws;                             // B*VIS*W
  float* relevant = visible + (size_t)B_ * VIS_ * W_;        // B*VIS
  float* isum = relevant + (size_t)B_ * VIS_;                // B*VIS
  float* scores = isum + (size_t)B_ * VIS_;                  // B*S*VIS
  float* ww = scores + (size_t)B_ * S_ * VIS_;               // B*S*VIS

  // 0: full copies (bandwidth-dominant: 256 MB => ~11 us at 23.3 TB/s)
  {
    int n4 = B_ * M_ * W_ / 4;
    int threads = (n4 + 1) / 2;
    k_copy4<<<(threads + 255) / 256, 256, 0, stream>>>((const v4f*)memory,
                                                       (v4f*)out_mem, n4);
    int u4 = B_ * M_ / 4;
    int uth = (u4 + 1) / 2;
    k_copy4<<<(uth + 255) / 256, 256, 0, stream>>>((const v4f*)usage,
                                                   (v4f*)out_usage, u4);
  }
  // 1: gather
  {
    int n = B_ * VIS_ * W_;
    k_gather<<<(n + 255) / 256, 256, 0, stream>>>(memory, usage, rp, visible,
                                                  relevant);
  }
  // 2: isum
  k_isum<<<(B_ * S_ + 255) / 256, 256, 0, stream>>>(relevant, isum);
  // 3: scores (WMMA f32)
  {
    dim3 g(VIS_ / 16 / 8, S_ / 16, B_);
    k_scores<<<g, 256, 0, stream>>>(keys, visible, scores);
  }
  // 4: softmax -> read_weights in place
  k_softmax<<<B_ * S_, 256, 0, stream>>>(scores);
  // 5: read_vectors (WMMA f32)
  {
    dim3 g(W_ / 16, S_ / 16, B_);
    k_readvec<<<g, 32, 0, stream>>>(scores, visible, out_rvec);
  }
  // 6: ww
  {
    size_t n = (size_t)B_ * S_ * VIS_;
    k_ww<<<(unsigned)((n + 255) / 256), 256, 0, stream>>>(scores, isum, ig, wg,
                                                          am, rp, ww);
  }
  // 7: activity + usage scatter
  k_usage<<<(B_ * VIS_ + 255) / 256, 256, 0, stream>>>(scores, ww, isum,
                                                       relevant, rp, out_usage);
  // 8: writings (WMMA f32) + erase + scatter into memory_new
  {
    dim3 g(VIS_ / 16, W_ / 16, B_);
    k_write<<<g, 32, 0, stream>>>(ww, wvec, visible, isum, rp, out_mem);
  }
  // 9: top-128 least used per batch
  k_topk<<<B_, 1024, 0, stream>>>(out_usage, out_lum);
}